// CenterLoss_50319836840009
// MI455X (gfx1250) — compile-verified
//
#include <hip/hip_runtime.h>
#include <stdint.h>

#define CL_BATCH 16384
#define CL_FEAT  256
#define CL_NCLS  100000

// ---------------- gfx1250 async global<->LDS helpers (inline CDNA5 asm) -----
// Generic pointers to __shared__ are {aperture_hi, lds_offset32}; the low 32
// bits are the wave-relative LDS byte address that VDST/VSRC expect.
__device__ __forceinline__ uint32_t lds_offset_of(const void* p) {
    return (uint32_t)(uintptr_t)p;
}

// copies 16 bytes from global address gaddr to LDS address lds_off   (tracked on ASYNCcnt)
__device__ __forceinline__ void async_load_lds_b128(uint32_t lds_off, const void* gaddr) {
    asm volatile("global_load_async_to_lds_b128 %0, %1, off"
                 :
                 : "v"(lds_off), "v"((uint64_t)(uintptr_t)gaddr)
                 : "memory");
}

// copies 4 bytes from LDS address lds_off to global address gaddr     (tracked on ASYNCcnt)
__device__ __forceinline__ void async_store_lds_b32(void* gaddr, uint32_t lds_off) {
    asm volatile("global_store_async_from_lds_b32 %0, %1, off"
                 :
                 : "v"((uint64_t)(uintptr_t)gaddr), "v"(lds_off)
                 : "memory");
}

__device__ __forceinline__ void wait_asynccnt0() {
    asm volatile("s_wait_asynccnt 0x0" ::: "memory");
}
// ----------------------------------------------------------------------------

// Kernel 1: new_centers = centers via the gfx1250 async-LDS DMA path, and zero
// the loss accumulator. B128 async loads (16B-aligned source) -> LDS -> four
// B32 async stores (output base is d_out+1 float, only 4B-aligned).
// Reads warm the 192MB L2 so kernel 2's random row gather hits L2.
__global__ void __launch_bounds__(256)
centerloss_copy_kernel(const float4* __restrict__ centers4,
                       float* __restrict__ out_centers,   // d_out + 1
                       float* __restrict__ out_loss) {
    __shared__ float stage[256 * 4];   // 4KB: one float4 slot per thread

    if (blockIdx.x == 0 && threadIdx.x == 0) {
        *out_loss = 0.0f;
    }

    // Exact mapping: 25000 blocks * 256 threads = 6,400,000 float4 chunks.
    const size_t j = (size_t)blockIdx.x * 256 + threadIdx.x;

    const uint32_t lds_off = lds_offset_of(&stage[threadIdx.x * 4]);

    // Async DMA: global -> LDS (16 bytes / lane)
    async_load_lds_b128(lds_off, centers4 + j);
    wait_asynccnt0();                       // LDS RAW: load done before store reads

    // Async DMA: LDS -> global (4 x 4 bytes / lane, 4B-aligned destination)
    float* o = out_centers + 4 * j;
    async_store_lds_b32(o + 0, lds_off + 0);
    async_store_lds_b32(o + 1, lds_off + 4);
    async_store_lds_b32(o + 2, lds_off + 8);
    async_store_lds_b32(o + 3, lds_off + 12);
    wait_asynccnt0();                       // drain before wave retires
}

// Kernel 2: per-sample gather, diff, fused (-lr*diff) scatter-add into the
// already-copied output, and block-reduced loss accumulation.
// Block = 256 threads = 8 wave32 = 4 samples x 64 threads (4 floats/thread).
__global__ void __launch_bounds__(256)
centerloss_scatter_kernel(const float4* __restrict__ x4,
                          const int*    __restrict__ labels,
                          const float4* __restrict__ centers4,
                          const float*  __restrict__ lr,
                          float* __restrict__ out_loss,
                          float* __restrict__ out_centers) {
    const int tid     = threadIdx.x;      // 0..255
    const int s_local = tid >> 6;         // 0..3  (sample within block)
    const int t       = tid & 63;         // 0..63 (float4 column within row)
    const int i       = blockIdx.x * 4 + s_local;   // sample index, grid exact

    const float neg_lr = -lr[0];

    const int lab = labels[i];
    const float4 xv = x4[(size_t)i   * (CL_FEAT / 4) + t];
    const float4 cv = centers4[(size_t)lab * (CL_FEAT / 4) + t];  // L2-resident row

    const float dx = cv.x - xv.x;
    const float dy = cv.y - xv.y;
    const float dz = cv.z - xv.z;
    const float dw = cv.w - xv.w;

    float partial = dx * dx + dy * dy + dz * dz + dw * dw;

    // Fused center update: new_centers[lab] -= lr * diff  (low-contention scatter)
    float* drow = out_centers + (size_t)lab * CL_FEAT + t * 4;
    atomicAdd(drow + 0, neg_lr * dx);
    atomicAdd(drow + 1, neg_lr * dy);
    atomicAdd(drow + 2, neg_lr * dz);
    atomicAdd(drow + 3, neg_lr * dw);

    // Loss reduction: wave32 shuffle tree -> LDS (8 waves) -> wave0 -> 1 atomic/block
    #pragma unroll
    for (int off = 16; off > 0; off >>= 1)
        partial += __shfl_down(partial, off, 32);

    __shared__ float red[8];
    const int wave = tid >> 5;
    const int lane = tid & 31;
    if (lane == 0) red[wave] = partial;
    __syncthreads();

    if (wave == 0) {
        float v = (lane < 8) ? red[lane] : 0.0f;
        #pragma unroll
        for (int off = 4; off > 0; off >>= 1)
            v += __shfl_down(v, off, 32);
        if (lane == 0)
            atomicAdd(out_loss, v * (0.5f / (float)CL_BATCH));
    }
}

extern "C" void kernel_launch(void* const* d_in, const int* in_sizes, int n_in,
                              void* d_out, int out_size, void* d_ws, size_t ws_size,
                              hipStream_t stream) {
    (void)in_sizes; (void)n_in; (void)d_ws; (void)ws_size; (void)out_size;

    const float4* x4       = (const float4*)d_in[0];   // [16384, 256] f32
    const int*    labels   = (const int*)   d_in[1];   // [16384] i32
    const float4* centers4 = (const float4*)d_in[2];   // [100000, 256] f32
    const float*  lr       = (const float*) d_in[3];   // [1] f32

    float* out        = (float*)d_out;
    float* out_loss   = out;        // out[0]
    float* out_center = out + 1;    // out[1 .. 25,600,000]

    // Kernel 1: async-DMA copy, exact grid: 6.4M float4 / 256 = 25000 blocks.
    centerloss_copy_kernel<<<25000, 256, 0, stream>>>(centers4, out_center, out_loss);

    // Kernel 2: 4096 blocks x 256 threads = 4 samples/block (stream-ordered).
    centerloss_scatter_kernel<<<CL_BATCH / 4, 256, 0, stream>>>(
        x4, labels, centers4, lr, out_loss, out_center);
}